// EdgeGNN_71365176590746
// MI455X (gfx1250) — compile-verified
//
#include <hip/hip_runtime.h>
#include <hip/hip_bf16.h>

typedef __attribute__((ext_vector_type(2))) float v2f;
typedef __attribute__((ext_vector_type(8))) float v8f;

#define D 64

// ---------------------------------------------------------------------------
// Kernel 1: fold W_n into the edge weights, storing PAIR-PACKED layouts:
//   value(k, j) of (W_n @ A)^T  is stored at  M1P[((k>>1)*64 + j)*2 + (k&1)]
// so a WMMA lane's two K-elements {B[k][n], B[k+1][n]} are contiguous 8 bytes
// in LDS -> single aligned ds_load_b64, no repacking movs.
//   c1[j] = sum_i W_n[j,i] * b_e[i]
// ---------------------------------------------------------------------------
__global__ void gnn_precompute(const float* __restrict__ W_e,
                               const float* __restrict__ b_e,
                               const float* __restrict__ W_n,
                               float* __restrict__ M1P,
                               float* __restrict__ M2P,
                               float* __restrict__ c1) {
    int tid = threadIdx.x;
    for (int idx = tid; idx < D * D; idx += blockDim.x) {
        int k = idx >> 6;
        int j = idx & 63;
        float s1 = 0.f, s2 = 0.f;
        for (int i = 0; i < D; ++i) {
            float wn = W_n[j * D + i];
            s1 += wn * W_e[i * (2 * D) + k];
            s2 += wn * W_e[i * (2 * D) + D + k];
        }
        int packed = (((k >> 1) * D) + j) * 2 + (k & 1);
        M1P[packed] = s1;
        M2P[packed] = s2;
    }
    if (tid < D) {
        float s = 0.f;
        for (int i = 0; i < D; ++i) s += W_n[tid * D + i] * b_e[i];
        c1[tid] = s;
    }
}

// ---------------------------------------------------------------------------
// Kernel 2: edge scatter.  One wave32 per edge: each lane carries 2 floats of
// the 64-float source row (coalesced 256B load, L2-resident: 12.8MB node
// table + 12.8MB accumulator both fit in the 192MB L2) and atomically
// accumulates into S[dst]. Lane 0 bumps the incoming-edge count.
// ---------------------------------------------------------------------------
__global__ void gnn_scatter(const float* __restrict__ x,
                            const long long* __restrict__ src,
                            const long long* __restrict__ dst,
                            float* __restrict__ S,
                            float* __restrict__ cnt,
                            int nEdges) {
    int edge = (int)((blockIdx.x * (unsigned)blockDim.x + threadIdx.x) >> 5);
    int lane = threadIdx.x & 31;
    if (edge >= nEdges) return;
    long long s = src[edge];
    long long d = dst[edge];
    const float2 v = *(const float2*)(x + s * D + lane * 2);
    float* p = S + d * D + lane * 2;
    atomicAdd(p, v.x);
    atomicAdd(p + 1, v.y);
    if (lane == 0) atomicAdd(cnt + d, 1.0f);
}

// ---------------------------------------------------------------------------
// Kernel 3: node update via fused fp32 WMMA GEMMs.
//   out[v,:] = (S[v,:]/max(c,1)) @ M1 + t_v * x[v,:] @ M2 + t_v*c1 + b_n
// One wave per 16 nodes; 4 N-tiles of 16 cols; K=64 as 16 x (16x16x4) WMMA
// steps per GEMM, both GEMMs accumulating into the same v8f accumulators.
// Pair-packed weights staged in LDS (33 KB of the 320 KB WGP LDS).
// ---------------------------------------------------------------------------
__global__ void __launch_bounds__(256)
gnn_node_update(const float* __restrict__ S,
                const float* __restrict__ cnt,
                const float* __restrict__ x,
                const float* __restrict__ M1P,
                const float* __restrict__ M2P,
                const float* __restrict__ c1g,
                const float* __restrict__ bng,
                float* __restrict__ out,
                int nGroups) {
    __shared__ alignas(16) float lM1[D * D];
    __shared__ alignas(16) float lM2[D * D];
    __shared__ float lc1[D];
    __shared__ float lbn[D];

    int tid = threadIdx.x;
    for (int i = tid; i < D * D; i += 256) {
        lM1[i] = M1P[i];
        lM2[i] = M2P[i];
    }
    if (tid < D) {
        lc1[tid] = c1g[tid];
        lbn[tid] = bng[tid];
    }
    __syncthreads();

    int wave  = tid >> 5;
    int lane  = tid & 31;
    int group = blockIdx.x * 8 + wave;   // wave-uniform
    if (group >= nGroups) return;        // uniform exit: EXEC all-1 for WMMA

    int base = group * 16;
    int rowA = lane & 15;   // A-operand row for this lane
    int hi   = lane >> 4;   // half-wave select (K sub-pair / M+8)

    float c   = cnt[base + rowA];
    float inv = 1.0f / fmaxf(c, 1.0f);   // 1/max(cnt,1)
    float t   = (c > 0.0f) ? 1.0f : 0.0f;

    const float* srow = S + (size_t)(base + rowA) * D;
    const float* xrow = x + (size_t)(base + rowA) * D;

    v8f acc0 = {}, acc1 = {}, acc2 = {}, acc3 = {};

    #pragma unroll
    for (int ks = 0; ks < 16; ++ks) {
        int kb = ks * 4 + hi * 2;        // this lane's K pair
        // A operands: mailbox-mean rows and gated node rows (contiguous b64)
        v2f am = { srow[kb] * inv, srow[kb + 1] * inv };
        v2f ax = { xrow[kb] * t,   xrow[kb + 1] * t   };
        int kp = (ks * 2 + hi) * D;      // pair-packed K-pair row base
        #pragma unroll
        for (int nt = 0; nt < 4; ++nt) {
            int ncol = nt * 16 + (lane & 15);
            // Single aligned 8B LDS load per B operand (pair-packed layout)
            v2f b1 = *(const v2f*)(lM1 + ((kp + ncol) << 1));
            v2f b2 = *(const v2f*)(lM2 + ((kp + ncol) << 1));
            v8f* accp = (nt == 0) ? &acc0 : (nt == 1) ? &acc1 : (nt == 2) ? &acc2 : &acc3;
            *accp = __builtin_amdgcn_wmma_f32_16x16x4_f32(
                false, am, false, b1, (short)0, *accp, false, false);
            *accp = __builtin_amdgcn_wmma_f32_16x16x4_f32(
                false, ax, false, b2, (short)0, *accp, false, false);
        }
    }

    // Per-output-row gate t[M]: rows live in lanes 0..15; hoisted shuffles.
    float tM[8];
    #pragma unroll
    for (int r = 0; r < 8; ++r) tM[r] = __shfl(t, r + 8 * hi, 32);

    // Bias + store. C/D layout: VGPR r -> M = r + 8*hi, N = lane&15 (+16*nt).
    #pragma unroll
    for (int nt = 0; nt < 4; ++nt) {
        int N = nt * 16 + (lane & 15);
        float bn  = lbn[N];
        float c1v = lc1[N];
        v8f acc = (nt == 0) ? acc0 : (nt == 1) ? acc1 : (nt == 2) ? acc2 : acc3;
        #pragma unroll
        for (int r = 0; r < 8; ++r) {
            int M = r + 8 * hi;
            out[(size_t)(base + M) * D + N] = acc[r] + bn + tM[r] * c1v;
        }
    }
}

// ---------------------------------------------------------------------------
extern "C" void kernel_launch(void* const* d_in, const int* in_sizes, int n_in,
                              void* d_out, int out_size, void* d_ws, size_t ws_size,
                              hipStream_t stream) {
    const float*     node = (const float*)d_in[0];      // [N, 64]
    const long long* src  = (const long long*)d_in[1];  // [E] int64
    const long long* dst  = (const long long*)d_in[2];  // [E] int64
    const float*     W_e  = (const float*)d_in[3];      // [64, 128]
    const float*     b_e  = (const float*)d_in[4];      // [64]
    const float*     W_n  = (const float*)d_in[5];      // [64, 64]
    const float*     b_n  = (const float*)d_in[6];      // [64]
    float*           out  = (float*)d_out;

    int nNodes = in_sizes[0] / D;
    int nEdges = in_sizes[1];

    // Workspace layout
    float* S    = (float*)d_ws;                 // nNodes*64
    float* cnt  = S + (size_t)nNodes * D;       // nNodes
    float* M1P  = cnt + nNodes;                 // 4096 (pair-packed)
    float* M2P  = M1P + D * D;                  // 4096 (pair-packed)
    float* c1   = M2P + D * D;                  // 64

    // Zero the accumulators (S + cnt) — memset node is graph-capture safe.
    hipMemsetAsync(S, 0, ((size_t)nNodes * D + nNodes) * sizeof(float), stream);

    gnn_precompute<<<1, 256, 0, stream>>>(W_e, b_e, W_n, M1P, M2P, c1);

    int edgesPerBlock = 256 / 32;
    int scatterBlocks = (nEdges + edgesPerBlock - 1) / edgesPerBlock;
    gnn_scatter<<<scatterBlocks, 256, 0, stream>>>(node, src, dst, S, cnt, nEdges);

    int nGroups = (nNodes + 15) / 16;           // 3125 (N is a multiple of 16)
    int nodeBlocks = (nGroups + 7) / 8;
    gnn_node_update<<<nodeBlocks, 256, 0, stream>>>(S, cnt, node, M1P, M2P, c1, b_n,
                                                    out, nGroups);
}